// SpatioTemporalModel_69604239999030
// MI455X (gfx1250) — compile-verified
//
#include <hip/hip_runtime.h>
#include <math.h>

typedef __attribute__((ext_vector_type(2))) float v2f;
typedef __attribute__((ext_vector_type(8))) float v8f;

#define FIN 16
#define HID 32

// ---------------------------------------------------------------- utilities
__global__ void zero_ws_kernel(float* __restrict__ p, long n) {
    long i = (long)blockIdx.x * blockDim.x + threadIdx.x;
    if (i < n) p[i] = 0.0f;
}

// deg[src[e]] += w[e]
__global__ void deg_kernel(const long long* __restrict__ src,
                           const float* __restrict__ w,
                           float* __restrict__ deg, int E) {
    int i = blockIdx.x * blockDim.x + threadIdx.x;
    if (i < E) atomicAdd(&deg[(int)src[i]], w[i]);
}

__global__ void dis_kernel(const float* __restrict__ deg,
                           float* __restrict__ dis, int N) {
    int i = blockIdx.x * blockDim.x + threadIdx.x;
    if (i < N) {
        float d = deg[i];
        dis[i] = (d > 0.0f) ? rsqrtf(fmaxf(d, 1e-12f)) : 0.0f;
    }
}

// tx1[dst,:] += (-dis[src]*w*dis[dst]) * x[src,:]
__global__ void scatter_kernel(const long long* __restrict__ src,
                               const long long* __restrict__ dst,
                               const float* __restrict__ w,
                               const float* __restrict__ dis,
                               const float* __restrict__ x,
                               float* __restrict__ tx1, int E) {
    int i = blockIdx.x * blockDim.x + threadIdx.x;
    if (i >= E) return;
    int s = (int)src[i];
    int d = (int)dst[i];
    float lw = -dis[s] * w[i] * dis[d];
    const float4* xr = reinterpret_cast<const float4*>(x + (size_t)s * FIN);
    float* o = tx1 + (size_t)d * FIN;
#pragma unroll
    for (int q = 0; q < 4; ++q) {
        float4 v = xr[q];
        atomicAdd(&o[q * 4 + 0], lw * v.x);
        atomicAdd(&o[q * 4 + 1], lw * v.y);
        atomicAdd(&o[q * 4 + 2], lw * v.z);
        atomicAdd(&o[q * 4 + 3], lw * v.w);
    }
}

// ---------------------------------------------------------------- gate GEMM
__device__ __forceinline__ v8f wmma4(v2f a, v2f b, v8f c) {
    // D = A(16x4,f32) * B(4x16,f32) + C(16x16,f32)
    return __builtin_amdgcn_wmma_f32_16x16x4_f32(
        /*neg_a=*/false, a, /*neg_b=*/false, b,
        /*c_mod=*/(short)0, c, /*reuse_a=*/false, /*reuse_b=*/false);
}

// One wave32 per 16-node tile: computes
//   Z  = sigmoid(x@Wz0 + tx1@Wz1 + bz)      (16x32)
//   Ht = tanh   (x@Wh0 + tx1@Wh1 + bh)      (16x32)
//   out = relu((1-Z)*Ht) @ Wlin + blin      (16x1)
__global__ void gate_kernel(const float* __restrict__ x,
                            const float* __restrict__ tx1,
                            const float* __restrict__ Wxz,   // (2,16,32)
                            const float* __restrict__ bxz,   // (32)
                            const float* __restrict__ bhz,   // (32)
                            const float* __restrict__ Wxh,   // (2,16,32)
                            const float* __restrict__ bxh,   // (32)
                            const float* __restrict__ bhh,   // (32)
                            const float* __restrict__ Wlin,  // (32,1)
                            const float* __restrict__ blin,  // (1)
                            float* __restrict__ out, int N, int numTiles) {
    const int lane = threadIdx.x & 31;
    const int wave = threadIdx.x >> 5;
    const int tile = blockIdx.x * (blockDim.x >> 5) + wave;
    if (tile >= numTiles) return;   // uniform per wave -> EXEC stays all-1s

    const int m0 = tile * 16;
    const int m  = lane & 15;       // A: row M ; B/D: column N (within half)
    const int hi = lane >> 4;       // lane-half selects K pair / M+8

    int row = m0 + m;
    if (row >= N) row = N - 1;      // clamp (N=100000 is a multiple of 16)
    const float* xrow = x   + (size_t)row * FIN;
    const float* trow = tx1 + (size_t)row * FIN;

    const float* Wz0 = Wxz;                 // K=0 term, row-major (16,32)
    const float* Wz1 = Wxz + FIN * HID;     // K=1 term
    const float* Wh0 = Wxh;
    const float* Wh1 = Wxh + FIN * HID;

    v8f accz0 = {}, accz1 = {}, acch0 = {}, acch1 = {};

    for (int k0 = 0; k0 < FIN; k0 += 4) {
        const int ka = k0 + 2 * hi;         // ISA A layout: V0/V1 = K pair,
                                            // lane halves = (K0,K1)/(K2,K3)
        v2f ax, at;
        ax[0] = xrow[ka]; ax[1] = xrow[ka + 1];
        at[0] = trow[ka]; at[1] = trow[ka + 1];

        auto ldb = [&](const float* W, int col) {
            v2f b;                          // B mirrors A: rows ka,ka+1, col n
            b[0] = W[ka * HID + col];
            b[1] = W[(ka + 1) * HID + col];
            return b;
        };

        accz0 = wmma4(ax, ldb(Wz0, m),      accz0);
        accz0 = wmma4(at, ldb(Wz1, m),      accz0);
        accz1 = wmma4(ax, ldb(Wz0, m + 16), accz1);
        accz1 = wmma4(at, ldb(Wz1, m + 16), accz1);
        acch0 = wmma4(ax, ldb(Wh0, m),      acch0);
        acch0 = wmma4(at, ldb(Wh1, m),      acch0);
        acch1 = wmma4(ax, ldb(Wh0, m + 16), acch1);
        acch1 = wmma4(at, ldb(Wh1, m + 16), acch1);
    }

    // per-lane column constants (n = m for half 0, m+16 for half 1)
    const float bz_0 = bxz[m]      + bhz[m];
    const float bz_1 = bxz[m + 16] + bhz[m + 16];
    const float bh_0 = bxh[m]      + bhh[m];
    const float bh_1 = bxh[m + 16] + bhh[m + 16];
    const float wl0  = Wlin[m];
    const float wl1  = Wlin[m + 16];
    const float bl   = blin[0];

#pragma unroll
    for (int r = 0; r < 8; ++r) {
        // D layout: VGPR r holds M=r (lanes 0-15) and M=r+8 (lanes 16-31)
        float z0 = 1.0f / (1.0f + expf(-(accz0[r] + bz_0)));
        float z1 = 1.0f / (1.0f + expf(-(accz1[r] + bz_1)));
        float t0 = tanhf(acch0[r] + bh_0);
        float t1 = tanhf(acch1[r] + bh_1);
        float h0 = (1.0f - z0) * t0; h0 = h0 > 0.0f ? h0 : 0.0f;
        float h1 = (1.0f - z1) * t1; h1 = h1 > 0.0f ? h1 : 0.0f;
        float v = h0 * wl0 + h1 * wl1;      // both N-halves folded in
        // reduce across the 16 lanes of each half (xor<16 keeps halves apart)
        v += __shfl_xor(v, 8);
        v += __shfl_xor(v, 4);
        v += __shfl_xor(v, 2);
        v += __shfl_xor(v, 1);
        int orow = m0 + r + 8 * hi;
        if (m == 0 && orow < N) out[orow] = v + bl;
    }
}

// ---------------------------------------------------------------- launcher
extern "C" void kernel_launch(void* const* d_in, const int* in_sizes, int n_in,
                              void* d_out, int out_size, void* d_ws, size_t ws_size,
                              hipStream_t stream) {
    const float*     x    = (const float*)d_in[0];
    const long long* ei   = (const long long*)d_in[1];  // int64 (2,E)
    const float*     ew   = (const float*)d_in[2];
    const float*     Wxz  = (const float*)d_in[3];
    const float*     bxz  = (const float*)d_in[4];
    const float*     bhz  = (const float*)d_in[6];
    // d_in[7..10] (W_xr,b_xr,W_hr,b_hr) and d_in[5],[13] unused: H==0 => R dead
    const float*     Wxh  = (const float*)d_in[11];
    const float*     bxh  = (const float*)d_in[12];
    const float*     bhh  = (const float*)d_in[14];
    const float*     Wlin = (const float*)d_in[15];
    const float*     blin = (const float*)d_in[16];
    float*           out  = (float*)d_out;

    const int N = in_sizes[0] / FIN;
    const int E = in_sizes[2];

    float* deg = (float*)d_ws;      // N
    float* dis = deg + N;           // N
    float* tx1 = dis + N;           // N*16

    const long zn = (long)N * 18;
    zero_ws_kernel<<<(int)((zn + 255) / 256), 256, 0, stream>>>((float*)d_ws, zn);

    deg_kernel<<<(E + 255) / 256, 256, 0, stream>>>(ei, ew, deg, E);
    dis_kernel<<<(N + 255) / 256, 256, 0, stream>>>(deg, dis, N);
    scatter_kernel<<<(E + 255) / 256, 256, 0, stream>>>(ei, ei + E, ew, dis, x, tx1, E);

    const int numTiles = (N + 15) / 16;
    const int wavesPerBlock = 8;                 // 256 threads = 8 wave32
    const int blocks = (numTiles + wavesPerBlock - 1) / wavesPerBlock;
    gate_kernel<<<blocks, 256, 0, stream>>>(x, tx1, Wxz, bxz, bhz,
                                            Wxh, bxh, bhh, Wlin, blin,
                                            out, N, numTiles);
}